// QuantumRNNCell_10943576670880
// MI455X (gfx1250) — compile-verified
//
#include <hip/hip_runtime.h>
#include <math.h>

#define NQ 4
#define KMAX 516              // D + NQ = 512 + 4
#define PI_F 3.14159265358979323846f

typedef __attribute__((ext_vector_type(2))) float v2f;
typedef __attribute__((ext_vector_type(8))) float v8f;

// ---------------------------------------------------------------------------
// Kernel 0: trig table for shared circuit params.
// trig[2*i] = cos(theta[i]/2), trig[2*i+1] = sin(theta[i]/2), i in [0,24)
// ---------------------------------------------------------------------------
__global__ void qrnn_trig(const float* __restrict__ theta,
                          float* __restrict__ trig) {
    int i = threadIdx.x;
    if (i < 2 * NQ * 3) {
        float h = theta[i] * 0.5f;
        trig[2 * i]     = cosf(h);
        trig[2 * i + 1] = sinf(h);
    }
}

// ---------------------------------------------------------------------------
// Kernel 1: angles = tanh([inputs, prev_h] @ W_enc + b_enc) * pi
// Wave-level WMMA f32 16x16x4. Each wave -> 16 rows; block of 256 -> 128 rows.
//
// A fragment (16x4 f32, ISA layout): lane L (m = L&15, g = L>>4):
//   VGPR0 = A[m][2g], VGPR1 = A[m][2g+1]       -> one float2 load / lane / step
// B fragment (4x16): VGPR0 = W[k0+2g][n], VGPR1 = W[k0+2g+1][n], n = L&15,
//   read from LDS copy of W_enc zero-padded to 16 columns.
// C tile (16x16): lane n<16: (M=j, N=n); lane n+16: (M=j+8, N=n). Cols 0..3 valid.
// ---------------------------------------------------------------------------
__global__ __launch_bounds__(256)
void qrnn_encode(const float* __restrict__ inputs,
                 const float* __restrict__ prev_h,
                 const float* __restrict__ W_enc,
                 const float* __restrict__ b_enc,
                 float* __restrict__ angles,
                 int B, int D) {
    __shared__ float wlds[KMAX * 16];          // 33 KB of the 320 KB WGP LDS

    const int K = D + NQ;                      // 516
    const int tid = threadIdx.x;

    // Stage W_enc -> LDS, zero-padded from 4 to 16 columns.
    for (int i = tid; i < K * 16; i += 256) {
        int r = i >> 4, c = i & 15;
        wlds[i] = (c < NQ) ? W_enc[r * NQ + c] : 0.0f;
    }
    __syncthreads();

    const int lane = tid & 31;
    const int wave = tid >> 5;
    const int g    = lane >> 4;                // which K-half this lane owns
    const int n    = lane & 15;                // A row / C column index
    const int rowBase = blockIdx.x * 128 + wave * 16;

    int rowA = rowBase + n;
    if (rowA >= B) rowA = B - 1;               // clamp (stores are predicated)

    const float* arow = inputs + (size_t)rowA * D;

    v8f acc = {};
    for (int k0 = 0; k0 < K; k0 += 4) {        // 129 uniform iterations
        const int ka = k0 + 2 * g;
        float2 av;
        if (k0 < D) {                          // uniform branch
            av = *(const float2*)(arow + ka);
        } else {                               // last step: prev_h columns
            av = *(const float2*)(prev_h + (size_t)rowA * NQ + (ka - D));
        }
        v2f a, b;
        a[0] = av.x;
        a[1] = av.y;
        b[0] = wlds[ka * 16 + n];
        b[1] = wlds[(ka + 1) * 16 + n];
        acc = __builtin_amdgcn_wmma_f32_16x16x4_f32(
            /*neg_a=*/false, a, /*neg_b=*/false, b,
            /*c_mod=*/(short)0, acc, /*reuse_a=*/false, /*reuse_b=*/false);
    }

    // Epilogue: cols 0..3 hold the 4 encoder outputs per row.
    if (n < NQ) {
        const float bias = b_enc[n];
#pragma unroll
        for (int j = 0; j < 8; ++j) {
            const int row = rowBase + j + g * 8;
            if (row < B) {
                angles[row * NQ + n] = tanhf(acc[j] + bias) * PI_F;
            }
        }
    }
}

// ---------------------------------------------------------------------------
// Kernel 2: 4-qubit statevector simulation + readout, one thread per row.
// Wire q <-> bit (3-q) of the 4-bit amplitude index (wire 0 = MSB).
// ---------------------------------------------------------------------------
__global__ __launch_bounds__(256)
void qrnn_circuit(const float* __restrict__ angles,
                  const float* __restrict__ trig,     // 48 floats
                  const float* __restrict__ W_out,    // (4,4) row-major
                  const float* __restrict__ b_out,    // (4,)
                  float* __restrict__ out,
                  int B) {
    __shared__ float s_trig[48];
    __shared__ float s_w[16];
    __shared__ float s_b[4];
    const int tid = threadIdx.x;
    if (tid < 48) s_trig[tid] = trig[tid];
    if (tid < 16) s_w[tid]    = W_out[tid];
    if (tid < 4)  s_b[tid]    = b_out[tid];
    __syncthreads();

    const int row = blockIdx.x * 256 + tid;
    if (row >= B) return;

    const float4 ang = *(const float4*)(angles + (size_t)row * NQ);
    float ch[NQ], sh[NQ];
    {
        const float* ap = &ang.x;
#pragma unroll
        for (int q = 0; q < NQ; ++q) {
            float h = ap[q] * 0.5f;
            ch[q] = cosf(h);
            sh[q] = sinf(h);
        }
    }

    // Product state after RY(angles) on |0000>: all-real amplitudes.
    float sr[16], si[16];
#pragma unroll
    for (int idx = 0; idx < 16; ++idx) {
        float v = ((idx >> 3) & 1) ? sh[0] : ch[0];
        v *= ((idx >> 2) & 1) ? sh[1] : ch[1];
        v *= ((idx >> 1) & 1) ? sh[2] : ch[2];
        v *= (idx & 1)        ? sh[3] : ch[3];
        sr[idx] = v;
        si[idx] = 0.0f;
    }

#pragma unroll
    for (int l = 0; l < 2; ++l) {
#pragma unroll
        for (int q = 0; q < NQ; ++q) {
            const float* t = s_trig + (l * NQ + q) * 6;
            const float ca = t[0], sa = t[1];
            const float cb = t[2], sb = t[3];
            const float cy = t[4], sy = t[5];
            const int w = 1 << (3 - q);
#pragma unroll
            for (int m = 0; m < 8; ++m) {
                const int i0 = ((m & ~(w - 1)) << 1) | (m & (w - 1));
                const int i1 = i0 | w;
                float r0 = sr[i0], x0 = si[i0];
                float r1 = sr[i1], x1 = si[i1];
                // RX(a): [[ca, -i sa], [-i sa, ca]]
                float ar0 = ca * r0 + sa * x1;
                float ai0 = ca * x0 - sa * r1;
                float ar1 = ca * r1 + sa * x0;
                float ai1 = ca * x1 - sa * r0;
                // RZ(b): amp0 *= (cb - i sb); amp1 *= (cb + i sb)
                float zr0 = cb * ar0 + sb * ai0;
                float zi0 = cb * ai0 - sb * ar0;
                float zr1 = cb * ar1 - sb * ai1;
                float zi1 = cb * ai1 + sb * ar1;
                // RY(c): [[cy, -sy], [sy, cy]] (real)
                sr[i0] = cy * zr0 - sy * zr1;
                si[i0] = cy * zi0 - sy * zi1;
                sr[i1] = sy * zr0 + cy * zr1;
                si[i1] = sy * zi0 + cy * zi1;
            }
        }
        // CNOT chain: control q (bit 3-q), target q+1 (bit 2-q)
#pragma unroll
        for (int q = 0; q < NQ - 1; ++q) {
            const int cm = 1 << (3 - q);
            const int tm = 1 << (2 - q);
#pragma unroll
            for (int idx = 0; idx < 16; ++idx) {
                if ((idx & cm) && !(idx & tm)) {
                    const int j = idx | tm;
                    float tr = sr[idx]; sr[idx] = sr[j]; sr[j] = tr;
                    float ti = si[idx]; si[idx] = si[j]; si[j] = ti;
                }
            }
        }
    }

    // PauliZ expectations per wire.
    float ev[NQ] = {0.0f, 0.0f, 0.0f, 0.0f};
#pragma unroll
    for (int idx = 0; idx < 16; ++idx) {
        const float p = sr[idx] * sr[idx] + si[idx] * si[idx];
        ev[0] += ((idx >> 3) & 1) ? -p : p;
        ev[1] += ((idx >> 2) & 1) ? -p : p;
        ev[2] += ((idx >> 1) & 1) ? -p : p;
        ev[3] += (idx & 1)        ? -p : p;
    }

    // next_h = tanh(ev @ W_out + b_out)
    float4 o;
    float* op = &o.x;
#pragma unroll
    for (int j = 0; j < NQ; ++j) {
        float acc = s_b[j];
#pragma unroll
        for (int q = 0; q < NQ; ++q) acc += ev[q] * s_w[q * NQ + j];
        op[j] = tanhf(acc);
    }
    *(float4*)(out + (size_t)row * NQ) = o;
}

// ---------------------------------------------------------------------------
extern "C" void kernel_launch(void* const* d_in, const int* in_sizes, int n_in,
                              void* d_out, int out_size, void* d_ws, size_t ws_size,
                              hipStream_t stream) {
    (void)n_in; (void)out_size; (void)ws_size;
    const float* inputs = (const float*)d_in[0];
    const float* prev_h = (const float*)d_in[1];
    const float* W_enc  = (const float*)d_in[2];
    const float* b_enc  = (const float*)d_in[3];
    const float* theta  = (const float*)d_in[4];
    const float* W_out  = (const float*)d_in[5];
    const float* b_out  = (const float*)d_in[6];
    float* out = (float*)d_out;

    const int B = in_sizes[1] / NQ;         // 32768
    const int D = in_sizes[0] / B;          // 512

    float* angles = (float*)d_ws;                       // B*4 floats
    float* trig   = angles + (size_t)B * NQ;            // 48 floats

    qrnn_trig<<<1, 32, 0, stream>>>(theta, trig);

    const int gridEnc = (B + 127) / 128;                // 128 rows / block
    qrnn_encode<<<gridEnc, 256, 0, stream>>>(inputs, prev_h, W_enc, b_enc,
                                             angles, B, D);

    qrnn_circuit<<<(B + 255) / 256, 256, 0, stream>>>(angles, trig, W_out,
                                                      b_out, out, B);
}